// gcn_classification_71150428225943
// MI455X (gfx1250) — compile-verified
//
#include <hip/hip_runtime.h>
#include <hip/hip_bf16.h>

#define HIDDEN 96
#define NCLS   10
#define NGRAPH 64

typedef float v2f __attribute__((ext_vector_type(2)));
typedef float v8f __attribute__((ext_vector_type(8)));

// ---------------------------------------------------------------------------
// fp32 WMMA GEMM: Out[N,96] = A[N,96] @ W[96,96]
// One wave computes one 16x16 output tile using V_WMMA_F32_16X16X4_F32,
// chaining K=96 as 24 k-slabs of 4.
//
// A (16x4 f32) lane layout: lanes 0-15 hold M=lane, K={0,1}; lanes 16-31 hold
// M=lane-16, K={2,3}.  B (4x16) mirrored: lanes 0-15 N=lane K={0,1}, lanes
// 16-31 N=lane-16 K={2,3}.  C/D (16x16 f32): VGPR r -> M=r (lanes 0-15) or
// M=r+8 (lanes 16-31), N = lane&15.
// ---------------------------------------------------------------------------
__global__ __launch_bounds__(256) void k_gemm96_wmma(const float* __restrict__ A,
                                                     const float* __restrict__ W,
                                                     float* __restrict__ Out,
                                                     int nTilesM) {
  const int tilesN = HIDDEN / 16;                       // 6
  int wave = blockIdx.x * (blockDim.x >> 5) + (threadIdx.x >> 5);
  int tM = wave / tilesN;
  int tN = wave - tM * tilesN;
  if (tM >= nTilesM) return;                            // wave-uniform exit

  int lane = threadIdx.x & 31;
  int half = lane >> 4;                                 // 0: K 0,1 | 1: K 2,3
  int l    = lane & 15;

  const float* __restrict__ arow = A + (size_t)(tM * 16 + l) * HIDDEN;
  const float* __restrict__ bcol = W + (tN * 16 + l);

  v8f acc = {};
#pragma unroll
  for (int kb = 0; kb < HIDDEN; kb += 4) {
    int ka = kb + half * 2;
    v2f a, b;
    a.x = arow[ka];
    a.y = arow[ka + 1];
    b.x = bcol[(size_t)ka * HIDDEN];
    b.y = bcol[(size_t)(ka + 1) * HIDDEN];
    // (neg_a, A, neg_b, B, c_mod, C, reuse_a, reuse_b)
    acc = __builtin_amdgcn_wmma_f32_16x16x4_f32(false, a, false, b,
                                                (short)0, acc, false, false);
  }

  float* __restrict__ orow =
      Out + (size_t)(tM * 16 + half * 8) * HIDDEN + tN * 16 + l;
#pragma unroll
  for (int r = 0; r < 8; ++r) orow[(size_t)r * HIDDEN] = acc[r];
}

// deg[dst] += 1 for every edge (deg buffer pre-zeroed)
__global__ void k_degree(const int* __restrict__ dst, float* __restrict__ deg, int E) {
  int e = blockIdx.x * blockDim.x + threadIdx.x;
  if (e < E) unsafeAtomicAdd(&deg[dst[e]], 1.0f);
}

// deg -> rsqrt(deg + 1) in place (self-loop included)
__global__ void k_dinv(float* __restrict__ d, int n) {
  int i = blockIdx.x * blockDim.x + threadIdx.x;
  if (i < n) d[i] = __frsqrt_rn(d[i] + 1.0f);
}

// norm[e] = dinv[src[e]] * dinv[dst[e]]
__global__ void k_edge_norm(const int* __restrict__ src, const int* __restrict__ dst,
                            const float* __restrict__ dinv, float* __restrict__ norm,
                            int E) {
  int e = blockIdx.x * blockDim.x + threadIdx.x;
  if (e < E) norm[e] = dinv[src[e]] * dinv[dst[e]];
}

// agg[dst*96 + f] += hW[src*96 + f] * norm[e]   (one thread per edge-feature)
__global__ void k_scatter(const float* __restrict__ hW, const int* __restrict__ src,
                          const int* __restrict__ dst, const float* __restrict__ norm,
                          float* __restrict__ agg, int total) {
  int i = blockIdx.x * blockDim.x + threadIdx.x;
  if (i >= total) return;
  int e = i / HIDDEN;
  int f = i - e * HIDDEN;
  float v = hW[(size_t)src[e] * HIDDEN + f] * norm[e];
  unsafeAtomicAdd(&agg[(size_t)dst[e] * HIDDEN + f], v);
}

// agg = maybe_relu(agg + hW * dinv^2 + bias)   in place
__global__ void k_self_bias_act(float* __restrict__ agg, const float* __restrict__ hW,
                                const float* __restrict__ dinv,
                                const float* __restrict__ bias, int total,
                                int do_relu) {
  int i = blockIdx.x * blockDim.x + threadIdx.x;
  if (i >= total) return;
  int node = i / HIDDEN;
  int f = i - node * HIDDEN;
  float di = dinv[node];
  float v = agg[i] + hW[i] * (di * di) + bias[f];
  agg[i] = do_relu ? fmaxf(v, 0.0f) : v;
}

// pooled sums over batch index + node counts (buffers pre-zeroed)
__global__ void k_pool(const float* __restrict__ h, const int* __restrict__ batch,
                       float* __restrict__ sums, float* __restrict__ cnts, int total) {
  int i = blockIdx.x * blockDim.x + threadIdx.x;
  if (i >= total) return;
  int node = i / HIDDEN;
  int f = i - node * HIDDEN;
  int g = batch[node];
  unsafeAtomicAdd(&sums[(size_t)g * HIDDEN + f], h[i]);
  if (f == 0) unsafeAtomicAdd(&cnts[g], 1.0f);
}

// out[g, c] = (sums[g] / max(cnt,1)) . W_lin[:, c] + b_lin[c]
__global__ void k_classifier(const float* __restrict__ sums, const float* __restrict__ cnts,
                             const float* __restrict__ Wl, const float* __restrict__ bl,
                             float* __restrict__ out) {
  int i = blockIdx.x * blockDim.x + threadIdx.x;
  if (i >= NGRAPH * NCLS) return;
  int g = i / NCLS;
  int c = i - g * NCLS;
  float inv = 1.0f / fmaxf(cnts[g], 1.0f);
  float acc = 0.0f;
#pragma unroll 4
  for (int f = 0; f < HIDDEN; ++f) acc += sums[g * HIDDEN + f] * Wl[f * NCLS + c];
  out[i] = acc * inv + bl[c];
}

extern "C" void kernel_launch(void* const* d_in, const int* in_sizes, int n_in,
                              void* d_out, int out_size, void* d_ws, size_t ws_size,
                              hipStream_t stream) {
  const float* x    = (const float*)d_in[0];
  const int*   ei   = (const int*)d_in[1];
  const int*   batch= (const int*)d_in[2];
  const float* W0   = (const float*)d_in[3];
  const float* b0   = (const float*)d_in[4];
  const float* W1   = (const float*)d_in[5];
  const float* b1   = (const float*)d_in[6];
  const float* W2   = (const float*)d_in[7];
  const float* b2   = (const float*)d_in[8];
  const float* Wlin = (const float*)d_in[9];
  const float* blin = (const float*)d_in[10];
  float* out = (float*)d_out;

  const int N = in_sizes[0] / HIDDEN;   // 50000
  const int E = in_sizes[1] / 2;        // 800000
  const int* src = ei;
  const int* dst = ei + E;

  // ---- workspace carve-up (floats) ----
  float* ws   = (float*)d_ws;
  size_t o    = 0;
  float* dinv = ws + o; o += ((size_t)N + 63) & ~(size_t)63;
  float* norm = ws + o; o += (size_t)E;
  float* sums = ws + o; o += (size_t)NGRAPH * HIDDEN;
  float* cnts = ws + o; o += 64;
  float* hW   = ws + o; o += (size_t)N * HIDDEN;
  float* aggA = ws + o; o += (size_t)N * HIDDEN;
  float* aggB = ws + o; o += (size_t)N * HIDDEN;
  (void)ws_size;

  const int nodeFeat = N * HIDDEN;            // 4.8M
  const int edgeFeat = E * HIDDEN;            // 76.8M
  const int TB = 256;
  const int nTilesM = N / 16;                 // 3125 (exact)
  const int gemmWaves = nTilesM * (HIDDEN / 16);
  const int gemmBlocks = (gemmWaves + 7) / 8; // 8 waves per 256-thread block

  // ---- degree / normalization (shared by all layers) ----
  hipMemsetAsync(dinv, 0, (size_t)N * sizeof(float), stream);
  k_degree<<<(E + TB - 1) / TB, TB, 0, stream>>>(dst, dinv, E);
  k_dinv<<<(N + TB - 1) / TB, TB, 0, stream>>>(dinv, N);
  k_edge_norm<<<(E + TB - 1) / TB, TB, 0, stream>>>(src, dst, dinv, norm, E);

  // ---- layer 0: in = x, out = aggA (ReLU) ----
  k_gemm96_wmma<<<gemmBlocks, TB, 0, stream>>>(x, W0, hW, nTilesM);
  hipMemsetAsync(aggA, 0, (size_t)nodeFeat * sizeof(float), stream);
  k_scatter<<<(edgeFeat + TB - 1) / TB, TB, 0, stream>>>(hW, src, dst, norm, aggA, edgeFeat);
  k_self_bias_act<<<(nodeFeat + TB - 1) / TB, TB, 0, stream>>>(aggA, hW, dinv, b0, nodeFeat, 1);

  // ---- layer 1: in = aggA, out = aggB (ReLU) ----
  k_gemm96_wmma<<<gemmBlocks, TB, 0, stream>>>(aggA, W1, hW, nTilesM);
  hipMemsetAsync(aggB, 0, (size_t)nodeFeat * sizeof(float), stream);
  k_scatter<<<(edgeFeat + TB - 1) / TB, TB, 0, stream>>>(hW, src, dst, norm, aggB, edgeFeat);
  k_self_bias_act<<<(nodeFeat + TB - 1) / TB, TB, 0, stream>>>(aggB, hW, dinv, b1, nodeFeat, 1);

  // ---- layer 2: in = aggB, out = aggA (no activation) ----
  k_gemm96_wmma<<<gemmBlocks, TB, 0, stream>>>(aggB, W2, hW, nTilesM);
  hipMemsetAsync(aggA, 0, (size_t)nodeFeat * sizeof(float), stream);
  k_scatter<<<(edgeFeat + TB - 1) / TB, TB, 0, stream>>>(hW, src, dst, norm, aggA, edgeFeat);
  k_self_bias_act<<<(nodeFeat + TB - 1) / TB, TB, 0, stream>>>(aggA, hW, dinv, b2, nodeFeat, 0);

  // ---- global mean pool + classifier ----
  hipMemsetAsync(sums, 0, (size_t)(NGRAPH * HIDDEN + 64) * sizeof(float), stream);
  k_pool<<<(nodeFeat + TB - 1) / TB, TB, 0, stream>>>(aggA, batch, sums, cnts, nodeFeat);
  k_classifier<<<(NGRAPH * NCLS + TB - 1) / TB, TB, 0, stream>>>(sums, cnts, Wlin, blin, out);
}